// Sub_Quantizer_29566554865869
// MI455X (gfx1250) — compile-verified
//
#include <hip/hip_runtime.h>
#include <hip/hip_bf16.h>

typedef float v2f __attribute__((ext_vector_type(2)));
typedef float v8f __attribute__((ext_vector_type(8)));
typedef int   v4i __attribute__((vector_size(16)));   // matches builtin param pointee

typedef __attribute__((address_space(1))) v4i* gptr_v4i;  // global
typedef __attribute__((address_space(3))) v4i* lptr_v4i;  // LDS

#define NQ    4
#define D     128
#define K     1024
#define Bsz   8
#define T     8192
#define TOK   512          // tokens per block
#define THREADS 512        // 16 wave32
#define PT    517          // resid LDS pitch (517 % 64 = 5, coprime -> conflict-free)
#define EPITCH 132         // embed tile pitch: 16B-aligned rows, bank stride 4
#define NKT   (K/16)       // 64 column tiles

#if defined(__HIP_DEVICE_COMPILE__) && __has_builtin(__builtin_amdgcn_global_load_async_to_lds_b128)
#define ASYNC_STAGE 1
#else
#define ASYNC_STAGE 0
#endif

static __device__ __forceinline__ v8f wmma_f32(v2f a, v2f b, v8f c) {
    // V_WMMA_F32_16X16X4_F32: D = A(16x4 f32) * B(4x16 f32) + C(16x16 f32)
    return __builtin_amdgcn_wmma_f32_16x16x4_f32(
        false, a, false, b, (short)0, c, false, false);
}

// stage one float4 (16B) of a 16x128 code tile into LDS
static __device__ __forceinline__ void stage_tile(const float* __restrict__ gsrc,
                                                  float* ldst) {
#if ASYNC_STAGE
    __builtin_amdgcn_global_load_async_to_lds_b128(
        (gptr_v4i)gsrc, (lptr_v4i)ldst, /*offset=*/0, /*cpol=*/0);
#else
    const float4 R = *(const float4*)gsrc;
    ((float2*)ldst)[0] = make_float2(R.x, R.y);
    ((float2*)ldst)[1] = make_float2(R.z, R.w);
#endif
}

static __device__ __forceinline__ void wait_stage() {
#if ASYNC_STAGE
  #if __has_builtin(__builtin_amdgcn_s_wait_asynccnt)
    __builtin_amdgcn_s_wait_asynccnt(0);
  #else
    asm volatile("s_wait_asynccnt 0x0" ::: "memory");
  #endif
#endif
}

// ---- Kernel 1: embed = scodebook[size]; esq[k] = ||embed[k]||^2 ----
__global__ void prep_embed_kernel(const float* __restrict__ scb,
                                  const long long* __restrict__ size64,
                                  float* __restrict__ embed,
                                  float* __restrict__ esq) {
    const int k = blockIdx.x;        // 1024 blocks
    const int d = threadIdx.x;       // 128 threads
    const long long row = size64[k];
    const float v = scb[(size_t)row * D + d];
    embed[(size_t)k * D + d] = v;
    float s = v * v;
    #pragma unroll
    for (int off = 16; off; off >>= 1) s += __shfl_xor(s, off, 32);
    __shared__ float part[4];
    if ((threadIdx.x & 31) == 0) part[threadIdx.x >> 5] = s;
    __syncthreads();
    if (threadIdx.x == 0) esq[k] = part[0] + part[1] + part[2] + part[3];
}

// ---- Kernel 2: fused RVQ (4 steps of GEMM + argmin + residual update) ----
__global__ void __launch_bounds__(THREADS, 1)
rvq_kernel(const float* __restrict__ z,
           const float* __restrict__ embed,
           const float* __restrict__ esq,
           const long long* __restrict__ size64,
           float* __restrict__ zq_out,
           float* __restrict__ code_out) {
    extern __shared__ float smem[];
    float* resid = smem;                        // [D][PT]  (d-major)
    float* ebuf  = resid + D * PT;              // 2 x [16][EPITCH]
    float* esq_s = ebuf + 2 * 16 * EPITCH;      // [K]
    int*   idx_s = (int*)(esq_s + K);           // [TOK]

    const int tid  = threadIdx.x;
    const int lane = tid & 31;
    const int wv   = tid >> 5;                  // 0..15
    const int l15  = lane & 15;
    const int hh   = lane >> 4;                 // 0: K/M low half, 1: high half
    const size_t g0 = (size_t)blockIdx.x * TOK; // global token base (within one b)
    const int bidx = (int)(g0 / T);
    const int t0   = (int)(g0 % T);
    const float* zb = z + (size_t)bidx * D * T;

    // stage e_sq
    for (int i = tid; i < K; i += THREADS) esq_s[i] = esq[i];

    // init residual from z (z is [B][D][T]; coalesced reads, contiguous LDS writes)
    #pragma unroll 4
    for (int d = 0; d < D; ++d)
        resid[d * PT + tid] = zb[(size_t)d * T + t0 + tid];
    __syncthreads();

    // embed tile staging roles: each thread owns one float4 of the 16x128 tile
    const int er  = tid >> 5;        // code row within tile 0..15
    const int ec4 = (tid & 31) * 4;  // dim column 0..124

    for (int step = 0; step < NQ; ++step) {
        // ---- load A operand (2 M-tiles x 32 k-blocks) in WMMA 16x4 f32 layout ----
        v2f a[2][32];
        #pragma unroll
        for (int tl = 0; tl < 2; ++tl) {
            const int tok = wv * 32 + tl * 16 + l15;
            #pragma unroll
            for (int kb = 0; kb < 32; ++kb) {
                const int d = kb * 4 + hh * 2;
                v2f av; av[0] = resid[d * PT + tok]; av[1] = resid[(d + 1) * PT + tok];
                a[tl][kb] = av;
            }
        }

        float bestv[2][8]; int besti[2][8];
        #pragma unroll
        for (int tl = 0; tl < 2; ++tl)
            #pragma unroll
            for (int i = 0; i < 8; ++i) { bestv[tl][i] = 3.4e38f; besti[tl][i] = 0; }

        // prologue: stage code tile 0 into buffer 0
        stage_tile(embed + (size_t)er * D + ec4,
                   ebuf + er * EPITCH + ec4);
        wait_stage();
        __syncthreads();

        for (int kt = 0; kt < NKT; ++kt) {
            // kick off DMA of next 16-code tile; overlaps with the WMMA block below
            if (kt + 1 < NKT)
                stage_tile(embed + (size_t)((kt + 1) * 16 + er) * D + ec4,
                           ebuf + ((kt + 1) & 1) * (16 * EPITCH) + er * EPITCH + ec4);

            const float* bb = ebuf + (kt & 1) * (16 * EPITCH);
            v8f c0 = {0,0,0,0,0,0,0,0};
            v8f c1 = {0,0,0,0,0,0,0,0};
            #pragma unroll
            for (int kb = 0; kb < 32; ++kb) {
                // B operand (4x16): lane -> code n=l15, dims 4kb+2h, +1
                const float2 bf = *(const float2*)(bb + l15 * EPITCH + kb * 4 + hh * 2);
                v2f bv; bv[0] = bf.x; bv[1] = bf.y;
                c0 = wmma_f32(a[0][kb], bv, c0);
                c1 = wmma_f32(a[1][kb], bv, c1);
            }

            // fused argmin update: score = ||e||^2 - 2*dot  (C: vgpr i -> M=i or i+8)
            const int kbase = kt * 16 + l15;
            const float es = esq_s[kbase];
            #pragma unroll
            for (int i = 0; i < 8; ++i) {
                const float s0 = es - 2.0f * c0[i];
                if (s0 < bestv[0][i]) { bestv[0][i] = s0; besti[0][i] = kbase; }
                const float s1 = es - 2.0f * c1[i];
                if (s1 < bestv[1][i]) { bestv[1][i] = s1; besti[1][i] = kbase; }
            }

            wait_stage();        // ASYNCcnt -> 0: next tile resident in LDS
            __syncthreads();     // single barrier per tile (reads were consumed above)
        }

        // ---- cross-lane argmin within each 16-lane half (C layout) ----
        #pragma unroll
        for (int tl = 0; tl < 2; ++tl) {
            #pragma unroll
            for (int i = 0; i < 8; ++i) {
                float v = bestv[tl][i]; int ix = besti[tl][i];
                #pragma unroll
                for (int off = 8; off; off >>= 1) {
                    const float ov = __shfl_xor(v, off, 16);
                    const int   oi = __shfl_xor(ix, off, 16);
                    if (ov < v || (ov == v && oi < ix)) { v = ov; ix = oi; }
                }
                if (l15 == 0)
                    idx_s[wv * 32 + tl * 16 + (hh << 3) + i] = ix;
            }
        }
        __syncthreads();

        // ---- residual -= embed[idx]; emit codes (size[idx]) ----
        const int dd = tid & 127;     // dim
        const int ts = tid >> 7;      // 0..3
        #pragma unroll 2
        for (int j = 0; j < TOK / 4; ++j) {
            const int t = j * 4 + ts;
            const int k = idx_s[t];
            const float val = embed[(size_t)k * D + dd];
            resid[dd * PT + t] -= val;
            if (dd == 0)
                code_out[(size_t)step * (Bsz * T) + g0 + t] = (float)size64[k];
        }
        __syncthreads();
    }

    // ---- zq = z - residual_final; transpose through LDS for coalesced output ----
    #pragma unroll 4
    for (int d = 0; d < D; ++d)
        resid[d * PT + tid] = zb[(size_t)d * T + t0 + tid] - resid[d * PT + tid];
    __syncthreads();
    {
        const int dd = tid & 127;
        const int ts = tid >> 7;
        #pragma unroll 2
        for (int j = 0; j < TOK / 4; ++j) {
            const int t = j * 4 + ts;
            zq_out[(g0 + t) * (size_t)D + dd] = resid[dd * PT + t];
        }
    }
}

extern "C" void kernel_launch(void* const* d_in, const int* in_sizes, int n_in,
                              void* d_out, int out_size, void* d_ws, size_t ws_size,
                              hipStream_t stream) {
    const float*     z    = (const float*)d_in[0];
    const float*     scb  = (const float*)d_in[1];
    const long long* sz64 = (const long long*)d_in[2];

    float* embed = (float*)d_ws;                 // [K][D] f32 = 512 KB
    float* esq   = embed + (size_t)K * D;        // [K]

    float* zq    = (float*)d_out;                // [B][T][D]
    float* codes = zq + (size_t)Bsz * T * D;     // [NQ][B][T] as float

    prep_embed_kernel<<<dim3(K), dim3(D), 0, stream>>>(scb, sz64, embed, esq);

    const size_t shmem = (size_t)(D * PT + 2 * 16 * EPITCH + K) * sizeof(float)
                       + (size_t)TOK * sizeof(int);   // ~281 KB (<=320 KB/WGP)
    rvq_kernel<<<dim3((Bsz * T) / TOK), dim3(THREADS), shmem, stream>>>(
        z, embed, esq, sz64, zq, codes);
}